// FM_47717086659306
// MI455X (gfx1250) — compile-verified
//
#include <hip/hip_runtime.h>
#include <hip/hip_bf16.h>

// Problem constants (reference: B=4, L1=L2=256, D=512)
#define BATCH   4
#define LSEQ    256
#define DDIM    512
#define MROWS   (BATCH * LSEQ)   // 1024 rows per embedding matrix
#define KCHUNK  128

typedef __attribute__((ext_vector_type(2))) float v2f;
typedef __attribute__((ext_vector_type(8))) float v8f;

// ---------------------------------------------------------------------------
// tanh: prefer gfx1250 v_tanh_f32, else v_exp_f32/v_rcp_f32 expansion.
// ---------------------------------------------------------------------------
__device__ __forceinline__ float tanh_fast(float x) {
#if __has_builtin(__builtin_amdgcn_tanhf)
    return __builtin_amdgcn_tanhf(x);
#elif __has_builtin(__builtin_amdgcn_exp2f) && __has_builtin(__builtin_amdgcn_rcpf)
    // tanh(x) = 1 - 2/(exp(2x)+1); exp(2x) = exp2(2*log2(e)*x)
    const float TWO_LOG2E = 2.885390081777927f;
    float e = __builtin_amdgcn_exp2f(x * TWO_LOG2E);
    return 1.0f - 2.0f * __builtin_amdgcn_rcpf(e + 1.0f);
#else
    return tanhf(x);
#endif
}

// ---------------------------------------------------------------------------
// Phase 1: H = X @ W (+bias), X:[1024,512] row-major, W:[512,512] row-major.
// One wave32 owns one 16x16 output tile; K swept 4 at a time with
// V_WMMA_F32_16X16X4_F32 (full fp32 precision).
//
// ISA VGPR layouts (CDNA5 ISA):
//   A 16x4 f32 : lane r=lane&15 is row M=r; half h=lane>>4: vgpr v holds K=v+2h
//   B 4x16 f32 : col = lane&15; vgpr v holds row K=v+2h
//   C/D 16x16  : vgpr v, half h -> row M = v + 8h; col = lane&15
// Grid is exact (2048 waves), so EXEC is all-ones for every WMMA.
// ---------------------------------------------------------------------------
__global__ __launch_bounds__(256) void fc1_half_wmma(
    const float* __restrict__ X,     // [MROWS, DDIM]
    const float* __restrict__ W,     // [DDIM, DDIM]  (K x N, row-major)
    const float* __restrict__ bias,  // [DDIM] or nullptr
    float* __restrict__ H)           // [MROWS, DDIM]
{
    const int lane = threadIdx.x & 31;
    const int wave = blockIdx.x * (blockDim.x >> 5) + (threadIdx.x >> 5);
    const int NT   = DDIM / 16;              // 32 tiles along N
    const int tm   = wave / NT;              // 0..63
    const int tn   = wave % NT;              // 0..31
    if (tm >= MROWS / 16) return;            // wave-uniform; keeps EXEC full

    const int r = lane & 15;                 // A row / B-D column within tile
    const int h = lane >> 4;                 // half-wave select

    const float* Arow = X + (tm * 16 + r) * DDIM + 2 * h;   // + k
    const float* Bptr = W + (2 * h) * DDIM + tn * 16 + r;   // + k*DDIM

    v8f c = {};
#pragma unroll 4
    for (int k = 0; k < DDIM; k += 4) {
        v2f a = *(const v2f*)(Arow + k);     // K = k+2h, k+2h+1 (8B aligned)
        v2f b;
        b.x = Bptr[(k    ) * DDIM];          // row K = k+2h
        b.y = Bptr[(k + 1) * DDIM];          // row K = k+2h+1
        c = __builtin_amdgcn_wmma_f32_16x16x4_f32(
                /*neg_a=*/false, a, /*neg_b=*/false, b,
                /*c_mod=*/(short)0, c, /*reuse_a=*/false, /*reuse_b=*/false);
    }

    const float bv = bias ? bias[tn * 16 + r] : 0.0f;  // fold b1 into h1
    float* Hout = H + (tm * 16 + 8 * h) * DDIM + tn * 16 + r;
#pragma unroll
    for (int v = 0; v < 8; ++v)
        Hout[v * DDIM] = c[v] + bv;          // row = v + 8h
}

// ---------------------------------------------------------------------------
// Phase 2 (dominant): out[b,i,j] = sum_k tanh(H1[b,i,k] + H2[b,j,k]) * W2[k] + b2
// Never materializes the [B,L1,L2,D] tensor (saves ~1 GB of HBM traffic).
// Block = 256 threads = (b, 16 i-rows, 32 j-cols); k chunked through LDS.
// s_h2 padded to stride KCHUNK+1 -> bank index (j+k)%64, conflict-free.
// ---------------------------------------------------------------------------
__global__ __launch_bounds__(256) void fused_tanh_gemv(
    const float* __restrict__ H1,   // [MROWS, DDIM], includes b1
    const float* __restrict__ H2,   // [MROWS, DDIM]
    const float* __restrict__ W2,   // [DDIM]
    const float* __restrict__ b2,   // [1]
    float* __restrict__ out)        // [BATCH, LSEQ, LSEQ]
{
    __shared__ __align__(16) float s_h1[16 * KCHUNK];        // 8 KB
    __shared__ __align__(16) float s_h2[32 * (KCHUNK + 1)];  // 16.5 KB
    __shared__ __align__(16) float s_w2[KCHUNK];             // 0.5 KB

    const int tid = threadIdx.x;
    const int bb  = blockIdx.z;
    const int i0  = blockIdx.y * 16;
    const int j0  = blockIdx.x * 32;
    const int i   = tid >> 5;     // 0..7 (also handles i+8)
    const int j   = tid & 31;     // 0..31 == lane -> padded-stride LDS reads

    const float* h1base = H1 + (bb * LSEQ + i0) * DDIM;
    const float* h2base = H2 + (bb * LSEQ + j0) * DDIM;

    float acc0 = 0.0f, acc1 = 0.0f;

    for (int k0 = 0; k0 < DDIM; k0 += KCHUNK) {
        // stage h1 tile: 16 x 128 = 512 float4s, coalesced
        for (int q = tid; q < 512; q += 256) {
            const int row = q >> 5, cf = q & 31;
            const float4 v = *(const float4*)(h1base + row * DDIM + k0 + cf * 4);
            *(float4*)(&s_h1[row * KCHUNK + cf * 4]) = v;
        }
        // stage h2 tile: 32 x 128 = 1024 float4s; scalar LDS stores (padded row)
        for (int q = tid; q < 1024; q += 256) {
            const int row = q >> 5, cf = q & 31;
            const float4 v = *(const float4*)(h2base + row * DDIM + k0 + cf * 4);
            float* d = &s_h2[row * (KCHUNK + 1) + cf * 4];
            d[0] = v.x; d[1] = v.y; d[2] = v.z; d[3] = v.w;
        }
        if (tid < KCHUNK) s_w2[tid] = W2[k0 + tid];
        __syncthreads();

        const float* p1a = &s_h1[i * KCHUNK];
        const float* p1b = &s_h1[(i + 8) * KCHUNK];
        const float* p2  = &s_h2[j * (KCHUNK + 1)];
#pragma unroll 8
        for (int k = 0; k < KCHUNK; ++k) {
            const float w  = s_w2[k];   // broadcast
            const float x2 = p2[k];     // conflict-free across lanes
            acc0 = __builtin_fmaf(tanh_fast(p1a[k] + x2), w, acc0);
            acc1 = __builtin_fmaf(tanh_fast(p1b[k] + x2), w, acc1);
        }
        __syncthreads();
    }

    const float bias2 = b2[0];
    float* o = out + (bb * LSEQ + i0) * LSEQ + (j0 + j);
    o[(i    ) * LSEQ] = acc0 + bias2;
    o[(i + 8) * LSEQ] = acc1 + bias2;
}

// ---------------------------------------------------------------------------
// Launch: inputs in setup_inputs() order:
//   0 emb_seq1 [4,256,512] f32   1 emb_seq2 [4,256,512] f32
//   2 W1 [1024,512] f32          3 b1 [512] f32
//   4 W2 [512,1] f32             5 b2 [1] f32
// Workspace: H1 (2 MB) + H2 (2 MB) in d_ws.
// ---------------------------------------------------------------------------
extern "C" void kernel_launch(void* const* d_in, const int* in_sizes, int n_in,
                              void* d_out, int out_size, void* d_ws, size_t ws_size,
                              hipStream_t stream) {
    const float* emb1 = (const float*)d_in[0];
    const float* emb2 = (const float*)d_in[1];
    const float* W1   = (const float*)d_in[2];
    const float* b1   = (const float*)d_in[3];
    const float* W2   = (const float*)d_in[4];
    const float* b2   = (const float*)d_in[5];

    float* H1 = (float*)d_ws;                 // [1024, 512]
    float* H2 = H1 + (size_t)MROWS * DDIM;    // [1024, 512]

    // Phase 1: 2048 tiles, 8 waves/block -> 256 blocks exactly (EXEC all-ones)
    dim3 blkA(256), grdA((MROWS / 16) * (DDIM / 16) / 8);
    fc1_half_wmma<<<grdA, blkA, 0, stream>>>(emb1, W1,                b1,      H1);
    fc1_half_wmma<<<grdA, blkA, 0, stream>>>(emb2, W1 + DDIM * DDIM,  nullptr, H2);

    // Phase 2: grid = (j-tiles, i-tiles, batch) = (8, 16, 4)
    dim3 blkB(256), grdB(LSEQ / 32, LSEQ / 16, BATCH);
    fused_tanh_gemv<<<grdB, blkB, 0, stream>>>(H1, H2, W2, b2, (float*)d_out);
}